// SoftMoERouter_12309376270336
// MI455X (gfx1250) — compile-verified
//
#include <hip/hip_runtime.h>
#include <hip/hip_bf16.h>
#include <math.h>

// ---------------------------------------------------------------------------
// SoftMoE router, MI455X (gfx1250, wave32).
// Dims: T=8192 tokens, D=2048, E=16 experts, H=1024, SL=1, NUM_SLOTS=16.
// Memory-bound (~450MB traffic, ~1.8 GFLOP). All skinny GEMMs use
// V_WMMA_F32_16X16X4_F32 (fp32 matrix cores). Deterministic reductions.
// ---------------------------------------------------------------------------

typedef float v2f __attribute__((ext_vector_type(2)));
typedef float v8f __attribute__((ext_vector_type(8)));

constexpr int T = 8192;      // B*S tokens
constexpr int D = 2048;
constexpr int E = 16;
constexpr int H = 1024;
constexpr int NS = 16;       // NUM_SLOTS
constexpr float EPSV = 1e-8f;

// ---- output layout (floats, concatenated in reference return order) -------
constexpr size_t OUT_OUT      = 0;                      // [T, D]
constexpr size_t OUT_RLOSS    = OUT_OUT + (size_t)T * D;       // 1
constexpr size_t OUT_GATE     = OUT_RLOSS + 1;                 // [T, E]
constexpr size_t OUT_TOPKI    = OUT_GATE + (size_t)T * E;      // [T, 2]
constexpr size_t OUT_TOPKW    = OUT_TOPKI + (size_t)T * 2;     // [T, 2]
constexpr size_t OUT_USAGE    = OUT_TOPKW + (size_t)T * 2;     // [E]
constexpr size_t OUT_ENT      = OUT_USAGE + E;                 // 1
constexpr size_t OUT_DISPATCH = OUT_ENT + 1;                   // [T, NS]

// ---- workspace layout (floats) --------------------------------------------
constexpr size_t WS_LOGITS      = 0;                          // [T, NS]
constexpr size_t WS_COLMAX      = WS_LOGITS + (size_t)T * NS; // [NS]
constexpr size_t WS_COLSUM      = WS_COLMAX + NS;             // [NS]
constexpr size_t WS_P           = WS_COLSUM + NS;             // [E]
constexpr size_t WS_PART        = WS_P + E;                   // [64][17]
constexpr size_t WS_SLOTIN_PART = WS_PART + 64 * 17;          // [16ks*128nt][256]
constexpr size_t WS_SLOTIN      = WS_SLOTIN_PART + (size_t)2048 * 256; // [NS, D]
constexpr size_t WS_HPART       = WS_SLOTIN + (size_t)NS * D; // [E*8][H]
constexpr size_t WS_H           = WS_HPART + (size_t)E * 8 * H;   // [E, H]
constexpr size_t WS_SOPART      = WS_H + (size_t)E * H;       // [E*4][D]
constexpr size_t WS_SLOTOUT     = WS_SOPART + (size_t)E * 4 * D;  // [NS, D]

static __device__ __forceinline__ v8f wmma_f32(v2f a, v2f b, v8f c) {
  // D = A(16x4 f32) * B(4x16 f32) + C(16x16 f32)
  return __builtin_amdgcn_wmma_f32_16x16x4_f32(false, a, false, b,
                                               (short)0, c, false, false);
}

// ============ K1: logits[T,16] = x[T,D] @ Wd[16,D]^T  (WMMA fp32) ==========
// One wave per 16-token tile; K=D in steps of 4.
__global__ __launch_bounds__(32) void k_logits(const float* __restrict__ x,
                                               const float* __restrict__ Wd,
                                               float* __restrict__ logits) {
  const int lane = threadIdx.x;
  const int m = lane & 15, g = lane >> 4;
  const int t0 = blockIdx.x * 16;
  const float* xrow = x + (size_t)(t0 + m) * D;   // A: row = token
  const float* wrow = Wd + (size_t)m * D;         // B: col n=m -> Wd row m
  v8f acc = {};
  for (int k0 = 0; k0 < D; k0 += 4) {
    const int ka = k0 + 2 * g;
    v2f a = *(const v2f*)(xrow + ka);
    v2f b = *(const v2f*)(wrow + ka);
    acc = wmma_f32(a, b, acc);
  }
#pragma unroll
  for (int r = 0; r < 8; ++r)
    logits[(size_t)(t0 + r + 8 * g) * NS + m] = acc[r];
}

// ============ K2: per-slot (column) softmax stats over T tokens ============
__global__ __launch_bounds__(256) void k_colstats(const float* __restrict__ logits,
                                                  float* __restrict__ colmax,
                                                  float* __restrict__ colsum) {
  __shared__ float sm[256];
  const int tid = threadIdx.x;
  const int c = tid & 15, r0 = tid >> 4;
  float mx = -INFINITY;
  for (int t = r0; t < T; t += 16) mx = fmaxf(mx, logits[(size_t)t * NS + c]);
  sm[tid] = mx;
  __syncthreads();
  for (int s = 8; s >= 1; s >>= 1) {
    if (r0 < s) sm[tid] = fmaxf(sm[tid], sm[(r0 + s) * 16 + c]);
    __syncthreads();
  }
  const float cmax = sm[c];
  __syncthreads();
  float sum = 0.f;
  for (int t = r0; t < T; t += 16) sum += __expf(logits[(size_t)t * NS + c] - cmax);
  sm[tid] = sum;
  __syncthreads();
  for (int s = 8; s >= 1; s >>= 1) {
    if (r0 < s) sm[tid] += sm[(r0 + s) * 16 + c];
    __syncthreads();
  }
  if (r0 == 0) { colmax[c] = cmax; colsum[c] = sm[c]; }
}

// ============ Kp: p = softmax over E of chunked means of Wd ================
// p_raw[e] = mean over slots s, d in [e*128, e*128+128) of Wd[s,d]
__global__ __launch_bounds__(128) void k_p(const float* __restrict__ Wd,
                                           float* __restrict__ p) {
  __shared__ float sm[128];
  __shared__ float praw[16];
  const int tid = threadIdx.x;
  const int e = tid >> 3, i = tid & 7;
  float s = 0.f;
  const int dbase = e * 128 + i * 16;
  for (int sl = 0; sl < NS; ++sl)
    for (int d = 0; d < 16; ++d) s += Wd[(size_t)sl * D + dbase + d];
  sm[tid] = s;
  __syncthreads();
  for (int st = 4; st >= 1; st >>= 1) {
    if (i < st) sm[tid] += sm[e * 8 + i + st];
    __syncthreads();
  }
  if (i == 0) praw[e] = sm[e * 8] / (16.f * 128.f);
  __syncthreads();
  if (tid == 0) {
    float mx = praw[0];
    for (int k = 1; k < 16; ++k) mx = fmaxf(mx, praw[k]);
    float ssum = 0.f, ex[16];
    for (int k = 0; k < 16; ++k) { ex[k] = __expf(praw[k] - mx); ssum += ex[k]; }
    for (int k = 0; k < 16; ++k) p[k] = ex[k] / ssum;
  }
}

// ============ K3: per-token combine/dispatch/top-k + block partials ========
__global__ __launch_bounds__(128) void k_route(const float* __restrict__ logits,
                                               const float* __restrict__ colmax,
                                               const float* __restrict__ colsum,
                                               float* __restrict__ out,
                                               float* __restrict__ partials) {
  __shared__ float sm[128 * 17];
  const int tid = threadIdx.x;
  const int t = blockIdx.x * 128 + tid;

  float l[16], comb[16];
#pragma unroll
  for (int e = 0; e < 16; ++e) l[e] = logits[(size_t)t * NS + e];

  float mx = l[0];
#pragma unroll
  for (int e = 1; e < 16; ++e) mx = fmaxf(mx, l[e]);
  float rsum = 0.f;
#pragma unroll
  for (int e = 0; e < 16; ++e) { comb[e] = __expf(l[e] - mx); rsum += comb[e]; }
  const float inv = 1.f / rsum;
#pragma unroll
  for (int e = 0; e < 16; ++e) {
    comb[e] *= inv;
    out[OUT_GATE + (size_t)t * NS + e] = comb[e];                       // gate_weights
    out[OUT_DISPATCH + (size_t)t * NS + e] =
        __expf(l[e] - colmax[e]) / colsum[e];                           // dispatch
  }

  // top-2 (first occurrence on ties, matching lax.top_k)
  int i1 = 0; float w1 = comb[0];
#pragma unroll
  for (int e = 1; e < 16; ++e) if (comb[e] > w1) { w1 = comb[e]; i1 = e; }
  int i2 = (i1 == 0) ? 1 : 0; float w2 = comb[i2];
#pragma unroll
  for (int e = 0; e < 16; ++e)
    if (e != i1 && comb[e] > w2) { w2 = comb[e]; i2 = e; }
  const float denom = fmaxf(w1 + w2, EPSV);
  out[OUT_TOPKI + (size_t)t * 2 + 0] = (float)i1;
  out[OUT_TOPKI + (size_t)t * 2 + 1] = (float)i2;
  out[OUT_TOPKW + (size_t)t * 2 + 0] = w1 / denom;
  out[OUT_TOPKW + (size_t)t * 2 + 1] = w2 / denom;

  float ent = 0.f;
#pragma unroll
  for (int e = 0; e < 16; ++e) ent -= comb[e] * __logf(comb[e] + EPSV);

#pragma unroll
  for (int e = 0; e < 16; ++e) sm[tid * 17 + e] = comb[e];
  sm[tid * 17 + 16] = ent;
  __syncthreads();
  for (int s = 64; s >= 1; s >>= 1) {
    if (tid < s)
      for (int k = 0; k < 17; ++k) sm[tid * 17 + k] += sm[(tid + s) * 17 + k];
    __syncthreads();
  }
  if (tid < 17) partials[blockIdx.x * 17 + tid] = sm[tid];
}

// ============ K3b: finalize mean_usage / entropy / router_loss =============
__global__ __launch_bounds__(32) void k_finalize(const float* __restrict__ partials,
                                                 const float* __restrict__ p,
                                                 float* __restrict__ out) {
  __shared__ float mu[16];
  const int tid = threadIdx.x;
  if (tid < 17) {
    float s = 0.f;
    for (int b = 0; b < 64; ++b) s += partials[b * 17 + tid];
    s /= (float)T;
    if (tid < 16) { out[OUT_USAGE + tid] = s; mu[tid] = s; }
    else out[OUT_ENT] = s;
  }
  __syncthreads();
  if (tid == 0) {
    float rl = 0.f;
    for (int e = 0; e < 16; ++e) rl += mu[e] * p[e];
    out[OUT_RLOSS] = (float)E * rl;
  }
}

// ============ K4: slot_in[16,D] = dispatch^T[16,T] @ x[T,D]  (WMMA fp32) ===
// grid = 128 n-tiles * 16 k-splits (512 tokens each), deterministic partials.
__global__ __launch_bounds__(32) void k_slotin_part(const float* __restrict__ x,
                                                    const float* __restrict__ dispatch,
                                                    float* __restrict__ part) {
  const int lane = threadIdx.x;
  const int m = lane & 15, g = lane >> 4;
  const int ntile = blockIdx.x & 127;
  const int ks = blockIdx.x >> 7;
  const int n0 = ntile * 16;
  const int kbase = ks * 512;
  v8f acc = {};
  for (int kt = kbase; kt < kbase + 512; kt += 4) {
    const int ka = kt + 2 * g;
    v2f a = { dispatch[(size_t)ka * NS + m], dispatch[(size_t)(ka + 1) * NS + m] };
    v2f b = { x[(size_t)ka * D + n0 + m],    x[(size_t)(ka + 1) * D + n0 + m] };
    acc = wmma_f32(a, b, acc);
  }
  float* pt = part + (size_t)blockIdx.x * 256;
#pragma unroll
  for (int r = 0; r < 8; ++r) pt[(r + 8 * g) * 16 + m] = acc[r];
}

__global__ __launch_bounds__(256) void k_slotin_reduce(const float* __restrict__ part,
                                                       float* __restrict__ slot_in) {
  const int idx = blockIdx.x * 256 + threadIdx.x;   // [0, 16*2048)
  const int d = idx & 2047;
  const int s = idx >> 11;
  const int ntile = d >> 4, nc = d & 15;
  float acc = 0.f;
  for (int ks = 0; ks < 16; ++ks)
    acc += part[(size_t)((ks << 7) | ntile) * 256 + s * 16 + nc];
  slot_in[idx] = acc;
}

// ============ K5: expert FFN (SL=1 -> GEMV), W coalesced, K-split ==========
__global__ __launch_bounds__(128) void k_ffn1_part(const float* __restrict__ slot_in,
                                                   const float* __restrict__ W1,
                                                   float* __restrict__ hpart) {
  __shared__ float xs[256];
  const int b = blockIdx.x;                 // 16 e * 8 jg * 8 ds
  const int e = b >> 6, rem = b & 63;
  const int jg = rem >> 3, ds = rem & 7;
  const int tid = threadIdx.x;
  const int j = jg * 128 + tid;
  const int d0 = ds * 256;
  xs[tid]       = slot_in[(size_t)e * D + d0 + tid];
  xs[tid + 128] = slot_in[(size_t)e * D + d0 + tid + 128];
  __syncthreads();
  const float* w = W1 + (size_t)e * D * H + (size_t)d0 * H + j;
  float acc = 0.f;
  for (int d = 0; d < 256; ++d) acc += xs[d] * w[(size_t)d * H];
  hpart[(size_t)(e * 8 + ds) * H + j] = acc;
}

__global__ __launch_bounds__(256) void k_ffn1_reduce(const float* __restrict__ hpart,
                                                     const float* __restrict__ b1,
                                                     float* __restrict__ h) {
  const int idx = blockIdx.x * 256 + threadIdx.x;   // [0, 16*1024)
  const int e = idx >> 10, j = idx & 1023;
  float v = b1[idx];
  for (int ds = 0; ds < 8; ++ds) v += hpart[(size_t)(e * 8 + ds) * H + j];
  h[idx] = 0.5f * v * (1.0f + erff(v * 0.70710678118654752f));  // exact gelu
}

__global__ __launch_bounds__(128) void k_ffn2_part(const float* __restrict__ h,
                                                   const float* __restrict__ W2,
                                                   float* __restrict__ sopart) {
  __shared__ float hs[256];
  const int b = blockIdx.x;                 // 16 e * 16 dg * 4 ks
  const int e = b >> 6, rem = b & 63;
  const int dg = rem >> 2, ks = rem & 3;
  const int tid = threadIdx.x;
  const int d = dg * 128 + tid;
  const int i0 = ks * 256;
  hs[tid]       = h[(size_t)e * H + i0 + tid];
  hs[tid + 128] = h[(size_t)e * H + i0 + tid + 128];
  __syncthreads();
  const float* w = W2 + (size_t)e * H * D + (size_t)i0 * D + d;
  float acc = 0.f;
  for (int i = 0; i < 256; ++i) acc += hs[i] * w[(size_t)i * D];
  sopart[(size_t)(e * 4 + ks) * D + d] = acc;
}

__global__ __launch_bounds__(256) void k_ffn2_reduce(const float* __restrict__ sopart,
                                                     const float* __restrict__ b2,
                                                     float* __restrict__ slot_out) {
  const int idx = blockIdx.x * 256 + threadIdx.x;   // [0, 16*2048)
  const int e = idx >> 11, d = idx & 2047;
  float v = b2[idx];
  for (int ks = 0; ks < 4; ++ks) v += sopart[(size_t)(e * 4 + ks) * D + d];
  slot_out[idx] = v;
}

// ============ K6: out[T,D] = combine[T,16] @ slot_out[16,D]  (WMMA fp32) ===
// One wave per 16-token tile x 128 columns (8 subtiles); A fragments reused.
__global__ __launch_bounds__(32) void k_out_wmma(const float* __restrict__ comb,
                                                 const float* __restrict__ slot_out,
                                                 float* __restrict__ out) {
  const int lane = threadIdx.x;
  const int m = lane & 15, g = lane >> 4;
  const int bm = blockIdx.x >> 4;   // 512 m-tiles
  const int bn = blockIdx.x & 15;   // 16 groups of 128 columns
  const int m0 = bm * 16;
  v2f a[4];
#pragma unroll
  for (int kk = 0; kk < 4; ++kk) {
    const int ka = kk * 4 + 2 * g;
    a[kk] = *(const v2f*)(comb + (size_t)(m0 + m) * NS + ka);
  }
#pragma unroll
  for (int sub = 0; sub < 8; ++sub) {
    const int n0 = bn * 128 + sub * 16;
    v8f acc = {};
#pragma unroll
    for (int kk = 0; kk < 4; ++kk) {
      const int ka = kk * 4 + 2 * g;
      v2f b = { slot_out[(size_t)ka * D + n0 + m],
                slot_out[(size_t)(ka + 1) * D + n0 + m] };
      acc = wmma_f32(a[kk], b, acc);
    }
#pragma unroll
    for (int r = 0; r < 8; ++r)
      out[(size_t)(m0 + r + 8 * g) * D + n0 + m] = acc[r];
  }
}

// ===========================================================================
extern "C" void kernel_launch(void* const* d_in, const int* in_sizes, int n_in,
                              void* d_out, int out_size, void* d_ws, size_t ws_size,
                              hipStream_t stream) {
  (void)in_sizes; (void)n_in; (void)out_size; (void)ws_size;
  const float* x  = (const float*)d_in[0];
  const float* Wd = (const float*)d_in[1];
  const float* W1 = (const float*)d_in[2];
  const float* b1 = (const float*)d_in[3];
  const float* W2 = (const float*)d_in[4];
  const float* b2 = (const float*)d_in[5];
  float* out = (float*)d_out;
  float* ws  = (float*)d_ws;

  k_logits      <<<T / 16, 32, 0, stream>>>(x, Wd, ws + WS_LOGITS);
  k_colstats    <<<1, 256, 0, stream>>>(ws + WS_LOGITS, ws + WS_COLMAX, ws + WS_COLSUM);
  k_p           <<<1, 128, 0, stream>>>(Wd, ws + WS_P);
  k_route       <<<64, 128, 0, stream>>>(ws + WS_LOGITS, ws + WS_COLMAX,
                                         ws + WS_COLSUM, out, ws + WS_PART);
  k_finalize    <<<1, 32, 0, stream>>>(ws + WS_PART, ws + WS_P, out);
  k_slotin_part <<<2048, 32, 0, stream>>>(x, out + OUT_DISPATCH, ws + WS_SLOTIN_PART);
  k_slotin_reduce<<<(NS * D) / 256, 256, 0, stream>>>(ws + WS_SLOTIN_PART, ws + WS_SLOTIN);
  k_ffn1_part   <<<1024, 128, 0, stream>>>(ws + WS_SLOTIN, W1, ws + WS_HPART);
  k_ffn1_reduce <<<(E * H) / 256, 256, 0, stream>>>(ws + WS_HPART, b1, ws + WS_H);
  k_ffn2_part   <<<1024, 128, 0, stream>>>(ws + WS_H, W2, ws + WS_SOPART);
  k_ffn2_reduce <<<(E * D) / 256, 256, 0, stream>>>(ws + WS_SOPART, b2, ws + WS_SLOTOUT);
  k_out_wmma    <<<(T / 16) * 16, 32, 0, stream>>>(out + OUT_GATE, ws + WS_SLOTOUT, out);
}